// PagedAttention_20925080666241
// MI455X (gfx1250) — compile-verified
//
#include <hip/hip_runtime.h>
#include <math.h>

#define B_      8
#define H_      32
#define D_      128
#define KVH_    8
#define L_      4
#define MAXS_   2048
#define R_      4                    // H / KVH  (GQA group)
#define NSPLIT_ 16
#define CHUNK_  (MAXS_ / NSPLIT_)    // 128 tokens per split
#define KSTR_   132                  // padded LDS row stride (floats); TDM pad reproduces this
#define TILEF_  (16 * KSTR_)         // floats per staged 16x128 tile

typedef __attribute__((ext_vector_type(2))) float    v2f;
typedef __attribute__((ext_vector_type(8))) float    v8f;
typedef __attribute__((ext_vector_type(4))) unsigned v4u;
typedef __attribute__((ext_vector_type(8))) int      v8i;
typedef __attribute__((ext_vector_type(4))) int      v4i;

#if __has_builtin(__builtin_amdgcn_tensor_load_to_lds) && __has_builtin(__builtin_amdgcn_s_wait_tensorcnt)
#define USE_TDM 1
#else
#define USE_TDM 0
#endif

// ---------------------------------------------------------------------------
// Pass 1: one wave32 per (split, kvh, b).  Flash-decoding partial attention
// for one 128-token chunk of one layer.  TDM gather-mode DMA pulls the paged
// 16x128 K/V tiles into LDS (double-buffered), fp32 WMMA 16x16x4 does QK^T
// and PV in transposed orientation.
// ---------------------------------------------------------------------------
__global__ __launch_bounds__(32)
void paged_attn_partial(const float* __restrict__ qsrc,   // [B,H,D] layer input
                        const float* __restrict__ kc,     // [NUM_SLOTS,L,KVH,D]
                        const float* __restrict__ vc,
                        const int*   __restrict__ slots,  // [B,MAXS]
                        const int*   __restrict__ lens,   // [B]
                        float* __restrict__ pO,           // [B*H, NSPLIT, D] unnormalized
                        float* __restrict__ pM,           // [B*H, NSPLIT]
                        float* __restrict__ pL,           // [B*H, NSPLIT]
                        int layer)
{
    __shared__ float kt[2 * TILEF_];    // double-buffered K tiles
    __shared__ float vt[2 * TILEF_];    // double-buffered V tiles
    __shared__ float qs[TILEF_];        // zero-padded 16-row Q tile (rows >= R are 0)

    const int lane  = threadIdx.x;
    const int split = blockIdx.x;
    const int kvh   = blockIdx.y;
    const int b     = blockIdx.z;
    const int half  = lane >> 4;        // wave half (0/1)
    const int n     = lane & 15;        // head column / s-row index

    // ---- stage zero-padded Q tile (kills per-iteration cndmask selects) ----
    for (int idx = lane; idx < TILEF_; idx += 32) qs[idx] = 0.0f;
    for (int idx = lane; idx < R_ * D_; idx += 32) {
        int r = idx >> 7, d = idx & (D_ - 1);
        qs[r * KSTR_ + d] = qsrc[((size_t)(b * H_) + (size_t)(kvh * R_ + r)) * D_ + d];
    }
    __syncthreads();

    const int   len   = lens[b];
    const int   cs    = split * CHUNK_;
    const float scale = 0.08838834764831845f;   // 1/sqrt(128)

    float mrun = -3.4e38f, lrun = 0.0f;
    v8f Oacc[8];
    for (int dt = 0; dt < 8; ++dt) Oacc[dt] = v8f{0,0,0,0,0,0,0,0};

#if USE_TDM
    // ---- static TDM descriptor group1: gather, 4B elems, 128-elem rows,
    //      row stride L*KVH*D=4096 elems, LDS pad 4 dwords per 128 dwords ----
    const v8i g1 = { (int)((2u << 16) | (1u << 20) | (6u << 22) | (3u << 25)),
                     (int)(128u   << 16),   // tensor_dim0 = 128        (bits 79:48)
                     (int)(16384u << 16),   // tensor_dim1 = NUM_SLOTS  (bits 111:80)
                     (int)(128u   << 16),   // tile_dim0   = 128        (bits 127:112)
                     16,                    // tile_dim1   = 16 gather indices
                     4096,                  // tensor_dim0_stride[31:0]
                     0, 0 };
    const v8i g4z = { 0, 0, 0, 0, 0, 0, 0, 0 };   // unused VADDR4 group (NULL)
    const uint64_t kga = (uint64_t)(uintptr_t)kc + (uint64_t)((layer * KVH_ + kvh) * D_) * 4u;
    const uint64_t vga = (uint64_t)(uintptr_t)vc + (uint64_t)((layer * KVH_ + kvh) * D_) * 4u;

    auto tdm_issue = [&](int t, int buf) {
        int slotv = slots[(size_t)b * MAXS_ + (cs + t * 16 + n)];
        int rs[8];
#pragma unroll
        for (int j = 0; j < 8; ++j) {
            int lo = __shfl(slotv, 2 * j,     32);
            int hi = __shfl(slotv, 2 * j + 1, 32);
            rs[j] = __builtin_amdgcn_readfirstlane(lo | (hi << 16));
        }
        v4i g2 = { rs[0], rs[1], rs[2], rs[3] };
        v4i g3 = { rs[4], rs[5], rs[6], rs[7] };
        unsigned kl = (unsigned)(uintptr_t)(kt + buf * TILEF_);
        unsigned vl = (unsigned)(uintptr_t)(vt + buf * TILEF_);
        v4u g0k = { 1u | (1u << 31), kl, (unsigned)kga, (unsigned)(kga >> 32) | (2u << 30) };
        v4u g0v = { 1u | (1u << 31), vl, (unsigned)vga, (unsigned)(vga >> 32) | (2u << 30) };
        __builtin_amdgcn_tensor_load_to_lds(g0k, g1, g2, g3, g4z, 0);
        __builtin_amdgcn_tensor_load_to_lds(g0v, g1, g2, g3, g4z, 0);
    };
#endif

    // ---- one 16-token tile: QK^T, online softmax, += P*V ----
    auto compute_tile = [&](const float* ktb, const float* vtb, int s0) {
        const int rv = (len - s0) < 16 ? (len - s0) : 16;
        v8f C = v8f{0,0,0,0,0,0,0,0};
        for (int d0 = 0; d0 < D_; d0 += 4) {
            float2 av = *(const float2*)(ktb + n * KSTR_ + d0 + half * 2);
            float2 qv = *(const float2*)(qs  + n * KSTR_ + d0 + half * 2);
            v2f a  = {av.x, av.y};
            v2f bm = {qv.x, qv.y};
            C = __builtin_amdgcn_wmma_f32_16x16x4_f32(false, a, false, bm,
                                                      (short)0, C, false, false);
        }
        float p[8];
        for (int j = 0; j < 8; ++j) {
            int rowj = j + 8 * half;
            float sc = C[j] * scale;
            p[j] = (rowj < rv) ? sc : -3.4e38f;
        }
        float tmax = p[0];
        for (int j = 1; j < 8; ++j) tmax = fmaxf(tmax, p[j]);
        tmax = fmaxf(tmax, __shfl_xor(tmax, 16, 32));
        float mnew  = fmaxf(mrun, tmax);
        float alpha = __expf(mrun - mnew);
        float lsum = 0.0f;
        for (int j = 0; j < 8; ++j) { p[j] = __expf(p[j] - mnew); lsum += p[j]; }
        lsum += __shfl_xor(lsum, 16, 32);
        lrun = lrun * alpha + lsum;
        mrun = mnew;
        for (int dt = 0; dt < 8; ++dt)
            for (int j = 0; j < 8; ++j) Oacc[dt][j] *= alpha;

        float sx[8];
        for (int j = 0; j < 8; ++j) sx[j] = __shfl_xor(p[j], 16, 32);

        for (int ks = 0; ks < 4; ++ks) {
            float b0, b1;
            if      (ks == 0) { b0 = half ? sx[2] : p[0]; b1 = half ? sx[3] : p[1]; }
            else if (ks == 1) { b0 = half ? sx[6] : p[4]; b1 = half ? sx[7] : p[5]; }
            else if (ks == 2) { b0 = half ? p[2] : sx[0]; b1 = half ? p[3] : sx[1]; }
            else              { b0 = half ? p[6] : sx[4]; b1 = half ? p[7] : sx[5]; }
            v2f bm = {b0, b1};
            const int srow = ks * 4 + half * 2;
            for (int dt = 0; dt < 8; ++dt) {
                v2f a = {vtb[srow * KSTR_ + dt * 16 + n],
                         vtb[(srow + 1) * KSTR_ + dt * 16 + n]};
                Oacc[dt] = __builtin_amdgcn_wmma_f32_16x16x4_f32(false, a, false, bm,
                                                                 (short)0, Oacc[dt],
                                                                 false, false);
            }
        }
    };

    const int avail = len - cs;
    if (avail > 0) {
        const int ntile = ((avail + 15) >> 4) < (CHUNK_ / 16) ? ((avail + 15) >> 4) : (CHUNK_ / 16);
#if USE_TDM
        tdm_issue(0, 0);
        for (int t = 0; t < ntile; ++t) {
            if (t + 1 < ntile) {
                // WAR guard: prior tile's LDS reads must land before DMA overwrites
                asm volatile("s_wait_dscnt 0" ::: "memory");
                tdm_issue(t + 1, (t + 1) & 1);
                __builtin_amdgcn_s_wait_tensorcnt(2);   // tile t resident (in-order)
            } else {
                __builtin_amdgcn_s_wait_tensorcnt(0);
            }
            compute_tile(kt + (t & 1) * TILEF_, vt + (t & 1) * TILEF_, cs + t * 16);
        }
#else
        for (int t = 0; t < ntile; ++t) {
            const int s0 = cs + t * 16;
            int slotv = slots[(size_t)b * MAXS_ + (s0 + n)];
            for (int i = 0; i < 16; ++i) {
                int sl = __shfl(slotv, i, 32);
                size_t base = (((size_t)sl * L_ + layer) * KVH_ + kvh) * (size_t)D_ + lane * 4;
                float4 kk = *(const float4*)(kc + base);
                float4 vv = *(const float4*)(vc + base);
                *(float4*)(kt + i * KSTR_ + lane * 4) = kk;
                *(float4*)(vt + i * KSTR_ + lane * 4) = vv;
            }
            __syncthreads();
            compute_tile(kt, vt, s0);
            __syncthreads();
        }
#endif
    }

    // ---- write partials for the R real heads (lane n and n+16 = d halves) ----
    if (n < R_) {
        size_t pbase = ((size_t)(b * H_ + kvh * R_ + n)) * NSPLIT_ + split;
        if (half == 0) { pM[pbase] = mrun; pL[pbase] = lrun; }
        float* od = pO + pbase * D_;
        for (int dt = 0; dt < 8; ++dt)
            for (int j = 0; j < 8; ++j)
                od[dt * 16 + j + 8 * half] = Oacc[dt][j];
    }
}

// ---------------------------------------------------------------------------
// Pass 2: combine NSPLIT partials per (b,h); one block per head, thread = d.
// ---------------------------------------------------------------------------
__global__ __launch_bounds__(128)
void paged_attn_combine(const float* __restrict__ pO, const float* __restrict__ pM,
                        const float* __restrict__ pL, float* __restrict__ dst)
{
    const int bh = blockIdx.x;
    const int d  = threadIdx.x;
    float M = -3.4e38f;
    for (int i = 0; i < NSPLIT_; ++i) M = fmaxf(M, pM[(size_t)bh * NSPLIT_ + i]);
    float acc = 0.0f, den = 0.0f;
    for (int i = 0; i < NSPLIT_; ++i) {
        float w = __expf(pM[(size_t)bh * NSPLIT_ + i] - M);
        den += w * pL[(size_t)bh * NSPLIT_ + i];
        acc += w * pO[((size_t)bh * NSPLIT_ + i) * D_ + d];
    }
    dst[(size_t)bh * D_ + d] = acc / den;
}

// ---------------------------------------------------------------------------
extern "C" void kernel_launch(void* const* d_in, const int* in_sizes, int n_in,
                              void* d_out, int out_size, void* d_ws, size_t ws_size,
                              hipStream_t stream)
{
    const float* query = (const float*)d_in[0];
    const float* kc    = (const float*)d_in[1];
    const float* vc    = (const float*)d_in[2];
    const int*   slots = (const int*)d_in[3];
    const int*   lens  = (const int*)d_in[4];
    float* out = (float*)d_out;

    float* pO = (float*)d_ws;                              // B*H*NSPLIT*D
    float* pM = pO + (size_t)B_ * H_ * NSPLIT_ * D_;       // B*H*NSPLIT
    float* pL = pM + (size_t)B_ * H_ * NSPLIT_;
    float* qb = pL + (size_t)B_ * H_ * NSPLIT_;            // B*H*D chained q

    for (int layer = 0; layer < L_; ++layer) {
        const float* qin  = (layer == 0) ? query : qb;
        float*       dstq = (layer == L_ - 1) ? out : qb;
        paged_attn_partial<<<dim3(NSPLIT_, KVH_, B_), 32, 0, stream>>>(
            qin, kc, vc, slots, lens, pO, pM, pL, layer);
        paged_attn_combine<<<dim3(B_ * H_), 128, 0, stream>>>(pO, pM, pL, dstq);
    }
}